// AssociativeMemory_81003083203014
// MI455X (gfx1250) — compile-verified
//
#include <hip/hip_runtime.h>
#include <hip/hip_bf16.h>

// AssociativeMemory for MI455X (gfx1250), wave32, WMMA f32 16x16x4.
// Bandwidth-bound on the 537MB attention output -> keep f32 precision,
// write attention exactly once, recompute logits (compute is ~free).

typedef __attribute__((ext_vector_type(2))) float v2f;
typedef __attribute__((ext_vector_type(8))) float v8f;

#define BATCH 4096
#define CAP   32768
#define KDIM  64

__device__ __forceinline__ v8f wmma4(v2f a, v2f b, v8f c) {
  // D = A(16x4) * B(4x16) + C, all f32
  return __builtin_amdgcn_wmma_f32_16x16x4_f32(
      /*neg_a=*/false, a, /*neg_b=*/false, b,
      /*c_mod=*/(short)0, c, /*reuse_a=*/false, /*reuse_b=*/false);
}

// -------- projections: out = tanh(X @ W^T + b), X:[4096,64] W:[64,64] -----
// grid 256 (16-row blocks), block 128 (4 waves, one 16-col tile each)
__global__ __launch_bounds__(128) void AM_proj(
    const float* __restrict__ X, const float* __restrict__ W,
    const float* __restrict__ bias, float* __restrict__ out) {
  const int tid = threadIdx.x, lane = tid & 31, wave = tid >> 5;
  const int rb = blockIdx.x;
  const int ln = lane & 15, h = lane >> 4, koff = h * 2;

  v2f a[16];
  const float* xrow = X + (size_t)(rb * 16 + ln) * KDIM;
#pragma unroll
  for (int ks = 0; ks < 16; ++ks) a[ks] = *(const v2f*)(xrow + ks * 4 + koff);

  v8f c = {};
  const float* wrow = W + (size_t)(wave * 16 + ln) * KDIM;
#pragma unroll
  for (int ks = 0; ks < 16; ++ks) {
    v2f b = *(const v2f*)(wrow + ks * 4 + koff);
    c = wmma4(a[ks], b, c);
  }
  const float bn = bias[wave * 16 + ln];
#pragma unroll
  for (int v = 0; v < 8; ++v)
    out[(size_t)(rb * 16 + v + 8 * h) * KDIM + wave * 16 + ln] = tanhf(c[v] + bn);
}

// -------- sim = KP @ MK^T, row argmax (ties -> lowest index) -------------
// grid 256, block 256 (8 waves); packed[row] = atomicMax of
// {monotonic(val)[63:32], ~idx[31:0]} so larger val wins, ties pick min idx.
__global__ __launch_bounds__(256) void AM_argmax(
    const float* __restrict__ KP, const float* __restrict__ MK,
    unsigned long long* __restrict__ packed) {
  const int tid = threadIdx.x, lane = tid & 31, wave = tid >> 5;
  const int rb = blockIdx.x;
  const int ln = lane & 15, h = lane >> 4, koff = h * 2;

  v2f a[16];
  const float* arow = KP + (size_t)(rb * 16 + ln) * KDIM;
#pragma unroll
  for (int ks = 0; ks < 16; ++ks) a[ks] = *(const v2f*)(arow + ks * 4 + koff);

  float best[8];
  int bidx[8];
#pragma unroll
  for (int v = 0; v < 8; ++v) { best[v] = -3.402823466e38f; bidx[v] = 0x7fffffff; }

  for (int ct = wave; ct < CAP / 16; ct += 8) {
    v8f c = {};
    const float* kb = MK + (size_t)(ct * 16 + ln) * KDIM;
#pragma unroll
    for (int ks = 0; ks < 16; ++ks) {
      v2f b = *(const v2f*)(kb + ks * 4 + koff);
      c = wmma4(a[ks], b, c);
    }
    const int col = ct * 16 + ln;
#pragma unroll
    for (int v = 0; v < 8; ++v) {
      float val = c[v];
      if (val > best[v] || (val == best[v] && col < bidx[v])) {
        best[v] = val; bidx[v] = col;
      }
    }
  }
  // reduce over the 16-lane group holding one row
#pragma unroll
  for (int v = 0; v < 8; ++v) {
    float bv = best[v]; int bi = bidx[v];
    for (int s = 1; s < 16; s <<= 1) {
      float ov = __shfl_xor(bv, s, 32);
      int oi = __shfl_xor(bi, s, 32);
      if (ov > bv || (ov == bv && oi < bi)) { bv = ov; bi = oi; }
    }
    best[v] = bv; bidx[v] = bi;
  }
  if (ln == 0) {
#pragma unroll
    for (int v = 0; v < 8; ++v) {
      unsigned ub = __float_as_uint(best[v]);
      ub = (ub & 0x80000000u) ? ~ub : (ub | 0x80000000u);
      unsigned long long pk =
          ((unsigned long long)ub << 32) | (unsigned)(~bidx[v]);
      atomicMax(&packed[rb * 16 + v + 8 * h], pk);
    }
  }
}

// -------- helpers -------------------------------------------------------
__global__ void AM_init(unsigned long long* packed, int* winner) {
  int i = blockIdx.x * 256 + threadIdx.x;
  if (i < BATCH) packed[i] = 0ull;
  if (i < CAP) winner[i] = -1;
}

__global__ void AM_winner(const unsigned long long* __restrict__ packed,
                          int* __restrict__ winner) {
  int b = blockIdx.x * 256 + threadIdx.x;
  if (b < BATCH) {
    int tm = (int)(~(unsigned)(packed[b] & 0xffffffffull));
    atomicMax(&winner[tm], b);
  }
}

__global__ void AM_build(const int* __restrict__ winner,
                         const float* __restrict__ KP, const float* __restrict__ SV,
                         const float* __restrict__ MK, const float* __restrict__ MV,
                         float* __restrict__ NK, float* __restrict__ NV) {
  int i = blockIdx.x * 256 + threadIdx.x;  // CAP*KDIM threads
  int slot = i >> 6, col = i & 63;
  int w = winner[slot];
  NK[i] = (w >= 0) ? KP[(size_t)w * KDIM + col] : MK[i];
  NV[i] = (w >= 0) ? SV[(size_t)w * KDIM + col] : MV[i];
}

// -------- attention + retrieved -----------------------------------------
// grid 256 (16 query rows each), block 256 (8 waves x 256 col tiles).
// 3 streaming passes over new_keys (8MB, L2-resident): max / sum+PV / write.
__global__ __launch_bounds__(256) void AM_attn(
    const float* __restrict__ Q, const float* __restrict__ NK,
    const float* __restrict__ NV, float* __restrict__ retrieved,
    float* __restrict__ attention) {
  __shared__ float red[8][16];
  __shared__ float rowmax[16];
  __shared__ float rowinv[16];
  __shared__ float rws[8][16 * 64];   // per-wave retrieved partials
  __shared__ float etile[8][16 * 16]; // per-wave E staging (C-layout -> A-layout)

  const int tid = threadIdx.x, lane = tid & 31, wave = tid >> 5;
  const int rb = blockIdx.x;
  const int ln = lane & 15, h = lane >> 4, koff = h * 2;

  v2f a[16];
  const float* qrow = Q + (size_t)(rb * 16 + ln) * KDIM;
#pragma unroll
  for (int ks = 0; ks < 16; ++ks) a[ks] = *(const v2f*)(qrow + ks * 4 + koff);

  // ---- pass 1: row max ----
  float mrow[8];
#pragma unroll
  for (int v = 0; v < 8; ++v) mrow[v] = -3.402823466e38f;
  for (int ct = wave; ct < CAP / 16; ct += 8) {
    v8f c = {};
    const float* kb = NK + (size_t)(ct * 16 + ln) * KDIM;
#pragma unroll
    for (int ks = 0; ks < 16; ++ks) {
      v2f b = *(const v2f*)(kb + ks * 4 + koff);
      c = wmma4(a[ks], b, c);
    }
#pragma unroll
    for (int v = 0; v < 8; ++v) mrow[v] = fmaxf(mrow[v], c[v]);
  }
#pragma unroll
  for (int v = 0; v < 8; ++v) {
    float x = mrow[v];
    for (int s = 1; s < 16; s <<= 1) x = fmaxf(x, __shfl_xor(x, s, 32));
    mrow[v] = x;
  }
  if (ln == 0) {
#pragma unroll
    for (int v = 0; v < 8; ++v) red[wave][v + 8 * h] = mrow[v];
  }
  __syncthreads();
  if (tid < 16) {
    float x = red[0][tid];
    for (int w = 1; w < 8; ++w) x = fmaxf(x, red[w][tid]);
    rowmax[tid] = x;
  }
  __syncthreads();

  // ---- pass 2: exp-sums + retrieved accumulation (E x V via WMMA) ----
  float rm[8], srow[8];
#pragma unroll
  for (int v = 0; v < 8; ++v) { rm[v] = rowmax[v + 8 * h]; srow[v] = 0.f; }
  v8f racc[4];
#pragma unroll
  for (int t = 0; t < 4; ++t) racc[t] = (v8f){};
  float* myet = &etile[wave][0];

  for (int ct = wave; ct < CAP / 16; ct += 8) {
    v8f c = {};
    const float* kb = NK + (size_t)(ct * 16 + ln) * KDIM;
#pragma unroll
    for (int ks = 0; ks < 16; ++ks) {
      v2f b = *(const v2f*)(kb + ks * 4 + koff);
      c = wmma4(a[ks], b, c);
    }
#pragma unroll
    for (int v = 0; v < 8; ++v) {
      float e = __expf(c[v] - rm[v]);
      srow[v] += e;
      myet[(v + 8 * h) * 16 + ln] = e;  // row-major 16x16 E tile
    }
    asm volatile("s_wait_dscnt 0" ::: "memory");  // wave-local LDS RAW order
#pragma unroll
    for (int ks = 0; ks < 4; ++ks) {  // K=16 slots in this col tile
      const int kbx = ks * 4 + koff;
      v2f ea;
      ea.x = myet[ln * 16 + kbx];
      ea.y = myet[ln * 16 + kbx + 1];
#pragma unroll
      for (int vt = 0; vt < 4; ++vt) {  // 4 x 16 value cols
        v2f bv;
        bv.x = NV[(size_t)(ct * 16 + kbx) * KDIM + vt * 16 + ln];
        bv.y = NV[(size_t)(ct * 16 + kbx + 1) * KDIM + vt * 16 + ln];
        racc[vt] = wmma4(ea, bv, racc[vt]);
      }
    }
  }
  // deterministic cross-wave reduction of sums and retrieved partials
#pragma unroll
  for (int v = 0; v < 8; ++v) {
    float x = srow[v];
    for (int s = 1; s < 16; s <<= 1) x += __shfl_xor(x, s, 32);
    srow[v] = x;
  }
  if (ln == 0) {
#pragma unroll
    for (int v = 0; v < 8; ++v) red[wave][v + 8 * h] = srow[v];
  }
#pragma unroll
  for (int vt = 0; vt < 4; ++vt)
#pragma unroll
    for (int v = 0; v < 8; ++v)
      rws[wave][(v + 8 * h) * 64 + vt * 16 + ln] = racc[vt][v];
  __syncthreads();
  if (tid < 16) {
    float x = 0.f;
    for (int w = 0; w < 8; ++w) x += red[w][tid];
    rowinv[tid] = 1.0f / x;
  }
  __syncthreads();
  for (int i = tid; i < 16 * 64; i += 256) {
    int row = i >> 6;
    float x = 0.f;
    for (int w = 0; w < 8; ++w) x += rws[w][i];
    retrieved[(size_t)(rb * 16 + row) * KDIM + (i & 63)] = x * rowinv[row];
  }

  // ---- pass 3: write normalized attention (single 537MB HBM write) ----
  float iv[8];
#pragma unroll
  for (int v = 0; v < 8; ++v) iv[v] = rowinv[v + 8 * h];
  for (int ct = wave; ct < CAP / 16; ct += 8) {
    v8f c = {};
    const float* kb = NK + (size_t)(ct * 16 + ln) * KDIM;
#pragma unroll
    for (int ks = 0; ks < 16; ++ks) {
      v2f b = *(const v2f*)(kb + ks * 4 + koff);
      c = wmma4(a[ks], b, c);
    }
#pragma unroll
    for (int v = 0; v < 8; ++v) {
      float av = __expf(c[v] - rm[v]) * iv[v];
      attention[(size_t)(rb * 16 + v + 8 * h) * CAP + ct * 16 + ln] = av;
    }
  }
}

extern "C" void kernel_launch(void* const* d_in, const int* in_sizes, int n_in,
                              void* d_out, int out_size, void* d_ws, size_t ws_size,
                              hipStream_t stream) {
  (void)in_sizes; (void)n_in; (void)out_size; (void)ws_size;
  const float* keys       = (const float*)d_in[0];
  const float* query      = (const float*)d_in[2];  // d_in[1] (values) unused, per reference
  const float* mem_keys   = (const float*)d_in[3];
  const float* mem_values = (const float*)d_in[4];
  const float* Wk = (const float*)d_in[5];
  const float* bk = (const float*)d_in[6];
  const float* Wv = (const float*)d_in[7];
  const float* bv = (const float*)d_in[8];

  char* ws = (char*)d_ws;
  float* kp = (float*)(ws + 0);                       // keys_proj   1MB
  float* qp = (float*)(ws + (1ull << 20));            // query_proj  1MB
  float* sv = (float*)(ws + (2ull << 20));            // stored_vals 1MB
  float* nk = (float*)(ws + (3ull << 20));            // new_keys    8MB
  float* nv = (float*)(ws + (11ull << 20));           // new_values  8MB
  unsigned long long* packed = (unsigned long long*)(ws + (19ull << 20));  // 32KB
  int* winner = (int*)(ws + (19ull << 20) + 65536);   // 128KB

  float* retrieved = (float*)d_out;
  float* attention = retrieved + (size_t)BATCH * KDIM;

  AM_init<<<CAP / 256, 256, 0, stream>>>(packed, winner);
  AM_proj<<<BATCH / 16, 128, 0, stream>>>(keys, Wk, bk, kp);
  AM_proj<<<BATCH / 16, 128, 0, stream>>>(query, Wk, bk, qp);
  AM_proj<<<BATCH / 16, 128, 0, stream>>>(query, Wv, bv, sv);
  AM_argmax<<<BATCH / 16, 256, 0, stream>>>(kp, mem_keys, packed);
  AM_winner<<<BATCH / 256, 256, 0, stream>>>(packed, winner);
  AM_build<<<(CAP * KDIM) / 256, 256, 0, stream>>>(winner, kp, sv, mem_keys,
                                                   mem_values, nk, nv);
  AM_attn<<<BATCH / 16, 256, 0, stream>>>(qp, nk, nv, retrieved, attention);
}